// OfdmSymbolRecovery_18133351923906
// MI455X (gfx1250) — compile-verified
//
#include <hip/hip_runtime.h>
#include <hip/hip_bf16.h>

typedef __attribute__((ext_vector_type(2))) float v2f;
typedef __attribute__((ext_vector_type(8))) float v8f;

#define IN_ROW_F   2560   // 2*(FFT_SIZE + CP_LEN)
#define OUT_ROW_F  1648   // 2*NSYMS
#define NSYMS      824
#define TWO_PI     6.283185307179586f
#define PI_F       3.14159265358979f

// D = A(16x4 f32) x B(4x16 f32) + C(16x16 f32), one tile per wave32.
__device__ __forceinline__ v8f wmma4(v2f a, v2f b, v8f c) {
    return __builtin_amdgcn_wmma_f32_16x16x4_f32(
        /*neg_a=*/false, a, /*neg_b=*/false, b,
        /*c_mod=*/(short)0, c, /*reuse_a=*/false, /*reuse_b=*/false);
}

__global__ __launch_bounds__(128)
void OfdmSymbolRecovery_kernel(const float* __restrict__ in,
                               float* __restrict__ out) {
    // LDS: 32x32 (+1 pad) complex work matrix, full corrected spectrum,
    // 32-entry W32 twiddle table, pilot scratch.
    __shared__ float xr[32 * 33];
    __shared__ float xi[32 * 33];
    __shared__ float s1r[1024];
    __shared__ float s1i[1024];
    __shared__ float tcos[32];
    __shared__ float tsin[32];
    __shared__ float ang[9];
    __shared__ float mn_s;

    const int tid  = threadIdx.x;
    const int lane = tid & 31;
    const int wv   = tid >> 5;      // 4 waves
    const int mt   = wv >> 1;       // tile row   (0..1) -> M base 0/16
    const int nt   = wv & 1;        // tile col   (0..1) -> N base 0/16
    const int lm   = lane & 15;     // lane's M/N within tile
    const int lh   = lane >> 4;     // K half-select

    const float* rowin = in + (size_t)blockIdx.x * IN_ROW_F;

    // ---- 32-entry twiddle table: e^{-2pi i p/32} = tcos[p] - i*tsin[p]
    if (tid < 32) {
        float s, c; __sincosf((float)tid * (TWO_PI / 32.0f), &s, &c);
        tcos[tid] = c;
        tsin[tid] = s;
    }

    // ---- load 1024 complex samples (offset CP_LEN/2 = 128 complex = 256 floats)
    // x[n1][n2] = sample[32*n1 + n2]
    for (int k = tid; k < 1024; k += 128) {
        float2 v = *(const float2*)(rowin + 256 + 2 * k);
        int n1 = k >> 5, n2 = k & 31;
        xr[n1 * 33 + n2] = v.x;
        xi[n1 * 33 + n2] = v.y;
    }
    __syncthreads();

    // ---- Stage A: Y = W32 * X   (Y[k1][n2] = sum_n1 e^{-2pi i k1 n1/32} x[n1][n2])
    v8f dr = {}, di = {};
    {
        const int M = 16 * mt + lm;           // k1 row of A/W32
        const int N = 16 * nt + lm;           // n2 col of B/X
        for (int kc = 0; kc < 8; ++kc) {
            v2f ar, ai, br, bi;
            #pragma unroll
            for (int v = 0; v < 2; ++v) {
                int K = 4 * kc + 2 * lh + v;  // n1
                int p = (M * K) & 31;
                ar[v] = tcos[p];
                ai[v] = -tsin[p];             // e^{-i th}
                br[v] = xr[K * 33 + N];
                bi[v] = xi[K * 33 + N];
            }
            v2f ain = -ai;
            dr = wmma4(ar,  br, dr);
            dr = wmma4(ain, bi, dr);
            di = wmma4(ar,  bi, di);
            di = wmma4(ai,  br, di);
        }
    }
    __syncthreads();   // all waves done reading X

    // ---- twiddle Z = Y * e^{-2pi i k1 n2 / 1024}, write back to LDS
    #pragma unroll
    for (int v = 0; v < 8; ++v) {
        int M = 16 * mt + v + 8 * lh;   // k1
        int N = 16 * nt + lm;           // n2
        float th = (float)((M * N) & 1023) * (TWO_PI / 1024.0f);
        float s, c; __sincosf(th, &s, &c);
        float zr = dr[v] * c + di[v] * s;
        float zi = di[v] * c - dr[v] * s;
        xr[M * 33 + N] = zr;
        xi[M * 33 + N] = zi;
    }
    __syncthreads();

    // ---- Stage C: Xf = Z * W32^T  (Xf[k1][k2] = sum_n2 Z[k1][n2] e^{-2pi i n2 k2/32})
    v8f er = {}, ei = {};
    {
        const int M = 16 * mt + lm;           // k1
        const int N = 16 * nt + lm;           // k2
        for (int kc = 0; kc < 8; ++kc) {
            v2f ar, ai, br, bi;
            #pragma unroll
            for (int v = 0; v < 2; ++v) {
                int K = 4 * kc + 2 * lh + v;  // n2
                ar[v] = xr[M * 33 + K];
                ai[v] = xi[M * 33 + K];
                int p = (K * N) & 31;
                br[v] = tcos[p];
                bi[v] = -tsin[p];
            }
            v2f ain = -ai;
            er = wmma4(ar,  br, er);
            er = wmma4(ain, bi, er);
            ei = wmma4(ar,  bi, ei);
            ei = wmma4(ai,  br, ei);
        }
    }

    // ---- writeback: fftshift roll + e^{+i pi t/4} ramp into s1[t]
    // cos(pi t/4) = tcos[(4t)&31], sin(pi t/4) = tsin[(4t)&31]
    #pragma unroll
    for (int v = 0; v < 8; ++v) {
        int k1 = 16 * mt + v + 8 * lh;
        int k2 = 16 * nt + lm;
        int k  = k1 + 32 * k2;          // spectrum index
        int t  = (k + 512) & 1023;      // rolled index
        int p  = (t << 2) & 31;
        float c = tcos[p];
        float s = tsin[p];
        s1r[t] = er[v] * c - ei[v] * s;
        s1i[t] = er[v] * s + ei[v] * c;
    }
    __syncthreads();

    // ---- pilot CFO estimate: 9 pilots at t = 95 + 100p
    if (tid < 9) {
        int t = 95 + 100 * tid;
        ang[tid] = atan2f(s1i[t], s1r[t]);
    }
    __syncthreads();
    if (tid == 0) {
        float acc = 0.0f;
        #pragma unroll
        for (int p = 0; p < 8; ++p) {
            float d = ang[p + 1] - ang[p];
            if (d >  PI_F) d -= 2.0f * PI_F;
            if (d < -PI_F) d += 2.0f * PI_F;
            acc += d;
        }
        mn_s = acc * (1.0f / 8.0f);
    }
    __syncthreads();

    // ---- de-rotate + gather 824 data subcarriers, interleaved re/im out
    const float mn = mn_s;
    float* rowout = out + (size_t)blockIdx.x * OUT_ROW_F;
    for (int m = tid; m < NSYMS; m += 128) {
        int t = (m < 792) ? (96 + (m / 99) * 100 + (m % 99)) : (m + 104);
        float ph = -(float)t * mn * 0.01f;   // -t*mn/PILOT_SPACING
        float s, c; __sincosf(ph, &s, &c);
        float2 o;
        o.x = s1r[t] * c - s1i[t] * s;
        o.y = s1r[t] * s + s1i[t] * c;
        *(float2*)(rowout + 2 * m) = o;
    }
}

extern "C" void kernel_launch(void* const* d_in, const int* in_sizes, int n_in,
                              void* d_out, int out_size, void* d_ws, size_t ws_size,
                              hipStream_t stream) {
    (void)n_in; (void)out_size; (void)d_ws; (void)ws_size;
    const float* in = (const float*)d_in[0];
    float* out = (float*)d_out;
    int rows = in_sizes[0] / IN_ROW_F;   // 16 * 1024 = 16384
    OfdmSymbolRecovery_kernel<<<rows, 128, 0, stream>>>(in, out);
}